// GINELayer_55843164783468
// MI455X (gfx1250) — compile-verified
//
#include <hip/hip_runtime.h>
#include <hip/hip_bf16.h>

typedef __attribute__((ext_vector_type(2))) float v2f;
typedef __attribute__((ext_vector_type(8))) float v8f;

// ---------------------------------------------------------------------------
// Zero-fill workspace accumulator (capture-safe, no hipMemsetAsync needed)
// ---------------------------------------------------------------------------
__global__ void gine_zero_f4(float4* __restrict__ p, int n4) {
    int i = blockIdx.x * blockDim.x + threadIdx.x;
    if (i < n4) p[i] = make_float4(0.f, 0.f, 0.f, 0.f);
}

// ---------------------------------------------------------------------------
// Edge kernel: one wave = 16 edges.
//   T = edge_attr[e0:e0+16, :] @ We            (WMMA f32 16x16x4, K=16, N=64)
//   msg = relu(T + be + x[src])                (fused epilogue)
//   atomicAdd(agg[dst], msg)                   (no-return f32 atomics -> L2)
// WMMA f32 layouts (wave32):
//   A 16x4 : lane L holds row (L&15), VGPR{0,1} = K = 2*(L>>4)+{0,1}
//   B 4x16 : lane L holds col (L&15), VGPR{0,1} = K = 2*(L>>4)+{0,1}
//   D 16x16: VGPR r, lane L -> (M = r + 8*(L>>4), N = L&15)
// ---------------------------------------------------------------------------
__global__ void gine_edge(const float* __restrict__ x,
                          const long long* __restrict__ edge_index, // [2,E] int64
                          const float* __restrict__ edge_attr,      // [E,16]
                          const float* __restrict__ We,             // [16,64]
                          const float* __restrict__ be,             // [64]
                          float* __restrict__ agg,                  // [N,64] (ws)
                          int E, int etiles) {
    const int lane = threadIdx.x & 31;
    const int wave = threadIdx.x >> 5;
    const int tile = blockIdx.x * (blockDim.x >> 5) + wave;
    if (tile >= etiles) return;                 // wave-uniform exit
    const int e0 = tile * 16;
    const int g  = lane >> 4;
    const int ln = lane & 15;

    // A operand: 4 K-steps of the 16x16 edge_attr tile
    v2f a[4];
    const float* arow = edge_attr + (size_t)(e0 + ln) * 16 + 2 * g;
    #pragma unroll
    for (int ks = 0; ks < 4; ++ks) a[ks] = *(const v2f*)(arow + ks * 4);

    // 4 N-tiles of 16 columns, K = 16 (4 WMMAs each)
    v8f acc[4];
    #pragma unroll
    for (int t = 0; t < 4; ++t) {
        v8f c = {};
        #pragma unroll
        for (int ks = 0; ks < 4; ++ks) {
            const float* bp = We + (size_t)(ks * 4 + 2 * g) * 64 + t * 16 + ln;
            v2f b; b.x = bp[0]; b.y = bp[64];
            c = __builtin_amdgcn_wmma_f32_16x16x4_f32(false, a[ks], false, b,
                                                      (short)0, c, false, false);
        }
        acc[t] = c;
    }

    const long long* srcp = edge_index;       // row 0: message source j
    const long long* dstp = edge_index + E;   // row 1: aggregation target i
    float bias[4];
    #pragma unroll
    for (int t = 0; t < 4; ++t) bias[t] = be[t * 16 + ln];

    #pragma unroll
    for (int r = 0; r < 8; ++r) {
        const int e = e0 + r + 8 * g;         // edge owned by this (lane, r)
        const int s = (int)srcp[e];
        const int d = (int)dstp[e];
        const float* xr = x + (size_t)s * 64;
        float* ar = agg + (size_t)d * 64;
        #pragma unroll
        for (int t = 0; t < 4; ++t) {
            const int n = t * 16 + ln;
            float v = acc[t][r] + bias[t] + xr[n];
            v = v > 0.f ? v : 0.f;
            atomicAdd(ar + n, v);             // global_atomic_add_f32 (no return)
        }
    }
}

// ---------------------------------------------------------------------------
// Node MLP kernel: one wave = 16 nodes.
//   h  = x + agg
//   r1 = relu(h @ W1 + b1)      (64 WMMAs; D-layout -> LDS tile)
//   out = r1 @ W2 + b2          (64 WMMAs; A re-read from LDS, transposed role)
// LDS row stride 68 floats: store banks (rows m and m+8 land on disjoint bank
// groups), load banks (68 % 64 == 4 -> 16 lanes hit 16 distinct banks).
// ---------------------------------------------------------------------------
__global__ void gine_mlp(const float* __restrict__ x,
                         const float* __restrict__ agg,
                         const float* __restrict__ W1,
                         const float* __restrict__ b1,
                         const float* __restrict__ W2,
                         const float* __restrict__ b2,
                         float* __restrict__ out,
                         int ntiles) {
    __shared__ float lds[8][16 * 68];
    const int lane = threadIdx.x & 31;
    const int wave = threadIdx.x >> 5;
    const int tile = blockIdx.x * (blockDim.x >> 5) + wave;
    if (tile >= ntiles) return;               // wave-uniform exit
    const int m0 = tile * 16;
    const int g  = lane >> 4;
    const int ln = lane & 15;
    float* ltile = &lds[wave][0];

    // A operand for GEMM1: h = x + agg, 16 K-steps
    v2f a[16];
    const float* xr = x   + (size_t)(m0 + ln) * 64 + 2 * g;
    const float* gr = agg + (size_t)(m0 + ln) * 64 + 2 * g;
    #pragma unroll
    for (int ks = 0; ks < 16; ++ks) {
        v2f xa = *(const v2f*)(xr + ks * 4);
        v2f ga = *(const v2f*)(gr + ks * 4);
        a[ks] = xa + ga;
    }

    // GEMM1 + bias + relu -> LDS (row-major 16x64, stride 68)
    #pragma unroll
    for (int t = 0; t < 4; ++t) {
        v8f c = {};
        #pragma unroll
        for (int ks = 0; ks < 16; ++ks) {
            const float* bp = W1 + (size_t)(ks * 4 + 2 * g) * 64 + t * 16 + ln;
            v2f b; b.x = bp[0]; b.y = bp[64];
            c = __builtin_amdgcn_wmma_f32_16x16x4_f32(false, a[ks], false, b,
                                                      (short)0, c, false, false);
        }
        const int n = t * 16 + ln;
        const float bias = b1[n];
        #pragma unroll
        for (int r = 0; r < 8; ++r) {
            float v = c[r] + bias;
            v = v > 0.f ? v : 0.f;
            ltile[(r + 8 * g) * 68 + n] = v;  // per-wave LDS: in-order, no barrier
        }
    }

    // A operand for GEMM2 from LDS (transposed role vs D layout)
    v2f a2[16];
    #pragma unroll
    for (int ks = 0; ks < 16; ++ks) {
        const float* lp = ltile + ln * 68 + ks * 4 + 2 * g;  // 8B aligned
        a2[ks] = *(const v2f*)lp;
    }

    // GEMM2 + bias -> out
    #pragma unroll
    for (int t = 0; t < 4; ++t) {
        v8f c = {};
        #pragma unroll
        for (int ks = 0; ks < 16; ++ks) {
            const float* bp = W2 + (size_t)(ks * 4 + 2 * g) * 64 + t * 16 + ln;
            v2f b; b.x = bp[0]; b.y = bp[64];
            c = __builtin_amdgcn_wmma_f32_16x16x4_f32(false, a2[ks], false, b,
                                                      (short)0, c, false, false);
        }
        const int n = t * 16 + ln;
        const float bias = b2[n];
        #pragma unroll
        for (int r = 0; r < 8; ++r)
            out[(size_t)(m0 + r + 8 * g) * 64 + n] = c[r] + bias;
    }
}

// ---------------------------------------------------------------------------
extern "C" void kernel_launch(void* const* d_in, const int* in_sizes, int n_in,
                              void* d_out, int out_size, void* d_ws, size_t ws_size,
                              hipStream_t stream) {
    const float*     x          = (const float*)d_in[0];
    const long long* edge_index = (const long long*)d_in[1];   // int64 [2,E]
    const float*     edge_attr  = (const float*)d_in[2];
    const float*     We         = (const float*)d_in[3];
    const float*     be         = (const float*)d_in[4];
    const float*     W1         = (const float*)d_in[5];
    const float*     b1         = (const float*)d_in[6];
    const float*     W2         = (const float*)d_in[7];
    const float*     b2         = (const float*)d_in[8];
    float*           out        = (float*)d_out;

    const int N = in_sizes[0] / 64;    // 50000
    const int E = in_sizes[2] / 16;    // 800000
    float* agg = (float*)d_ws;         // [N,64] f32 accumulator in workspace

    // 1) zero the scatter accumulator
    const int n4 = (N * 64) / 4;
    gine_zero_f4<<<(n4 + 255) / 256, 256, 0, stream>>>((float4*)agg, n4);

    // 2) per-edge transform + gather + relu + atomic scatter
    const int etiles = (E + 15) / 16;
    gine_edge<<<(etiles + 7) / 8, 256, 0, stream>>>(x, edge_index, edge_attr,
                                                    We, be, agg, E, etiles);

    // 3) node MLP (two WMMA GEMMs, fused bias/relu)
    const int ntiles = (N + 15) / 16;
    gine_mlp<<<(ntiles + 7) / 8, 256, 0, stream>>>(x, agg, W1, b1, W2, b2,
                                                   out, ntiles);
}